// Inception1_31447750542144
// MI455X (gfx1250) — compile-verified
//
#include <hip/hip_runtime.h>

// Problem constants (fixed by the reference).
constexpr int B_    = 8192;
constexpr int H_    = 9;
constexpr int W_    = 240;
constexpr int S_    = 10;
constexpr int T_    = 24;    // W/S
constexpr int NW_   = 8;     // T/3 pooled windows
constexpr int CTOT  = 144;   // 36+36+9+9+9+9+36
constexpr int HT_   = H_ * T_;        // 216
constexpr int ROW_  = H_ * W_;        // 2160 floats per batch row
constexpr int OUTB  = 3 * CTOT * NW_; // 3456 outputs per batch row

typedef float v2f __attribute__((ext_vector_type(2)));
typedef float v8f __attribute__((ext_vector_type(8)));
typedef int   v4i __attribute__((vector_size(16)));
typedef __attribute__((address_space(1))) v4i* gptr_v4i;
typedef __attribute__((address_space(3))) v4i* lptr_v4i;

#if defined(__has_builtin)
#if __has_builtin(__builtin_amdgcn_global_load_async_to_lds_b128) && \
    __has_builtin(__builtin_amdgcn_s_wait_asynccnt)
#define USE_ASYNC_LDS 1
#endif
#endif

__device__ __forceinline__ float fsqrt_hw(float x) {
#if __has_builtin(__builtin_amdgcn_sqrtf)
    return __builtin_amdgcn_sqrtf(x);
#else
    return sqrtf(x);
#endif
}
__device__ __forceinline__ float frcp_hw(float x) {
#if __has_builtin(__builtin_amdgcn_rcpf)
    return __builtin_amdgcn_rcpf(x);
#else
    return 1.0f / x;
#endif
}

__device__ __forceinline__ int grp_of(int c) {
    return (c < 36) ? 0 : (c < 72) ? 1 : (c < 81) ? 2 : (c < 90) ? 3
         : (c < 99) ? 4 : (c < 108) ? 5 : 6;
}

// ---------------- Pass 0: mean table + zero accumulators -------------------
__global__ void mean_kernel(const float* __restrict__ data1,
                            float* __restrict__ meantab,
                            float* __restrict__ gacc) {
    int tid = blockIdx.x * 256 + threadIdx.x;
    if (tid < 64) gacc[tid] = 0.0f;
    if (tid < B_ * HT_) {
        int b = tid / HT_;
        int r = tid - b * HT_;
        int h = r / T_;
        int t = r - h * T_;
        const float* p = data1 + (size_t)b * ROW_ + h * W_ + t * S_;
        float s = 0.0f;
#pragma unroll
        for (int k = 0; k < S_; ++k) s += p[k];
        meantab[tid] = s * 0.1f;
    }
}

// ---------------- Pass 1 (STATS) / Pass 3 (OUTPUT) -------------------------
// One wave (32 lanes) per batch row. Per t: Gram G = M*M^T via 3 chained
// V_WMMA_F32_16X16X4_F32 (M = [x0..x8; dw; ones], 16x12 f32), staged through
// LDS; lanes then cover the 144 feature channels (5 strided slots each).
template <bool STATS>
__global__ __launch_bounds__(32)
void feat_kernel(const float* __restrict__ data1,
                 const int*   __restrict__ pi,
                 const int*   __restrict__ pj,
                 const float* __restrict__ meantab,
                 float*       __restrict__ gacc,    // STATS only
                 const float* __restrict__ params,  // OUTPUT only (7x8 floats)
                 float*       __restrict__ out) {
    __shared__ __align__(16) float row[ROW_];
    __shared__ float G[256];
    __shared__ float red[64];

    const int b    = blockIdx.x;
    const int lane = threadIdx.x;
    const int r    = lane & 15;
    const int hv   = lane >> 4;

    // ---- Stage the whole 2160-float batch row into LDS (async DMA path) ----
    {
        const float* srcb = data1 + (size_t)b * ROW_;
#ifdef USE_ASYNC_LDS
        for (int i = lane; i < ROW_ / 4; i += 32) {
            __builtin_amdgcn_global_load_async_to_lds_b128(
                (gptr_v4i)(float*)(srcb + 4 * i),
                (lptr_v4i)(&row[4 * i]),
                0, 0);
        }
        __builtin_amdgcn_s_wait_asynccnt(0);
#else
        const float4* src = (const float4*)srcb;
        float4* dst = (float4*)row;
        for (int i = lane; i < ROW_ / 4; i += 32) dst[i] = src[i];
#endif
    }
    if (STATS) {
        for (int i = lane; i < 64; i += 32) red[i] = 0.0f;
    } else {
        for (int i = lane; i < 64; i += 32) red[i] = params[i];
    }
    __syncthreads();

    // ---- Per-lane WMMA operand descriptors (t-invariant) ----
    // Element j (0..5): step s=j/2, elem e=j%2, K = 4*s + 2*hv + e.
    // Branchless fetch: each value is fsel times the LDS row entry at moff, plus cval (cval==0 for data elems).
    float fsel[6], cval[6];
    int   moff[6];
#pragma unroll
    for (int j = 0; j < 6; ++j) {
        int k = 4 * (j / 2) + 2 * hv + (j & 1);
        bool id = (r < 9) && (k < S_);
        fsel[j] = id ? 1.0f : 0.0f;
        cval[j] = (k < S_) ? ((r == 9) ? (float)(k + 1) * (1.0f / 55.0f)
                                       : (r == 10 ? 1.0f : 0.0f))
                           : 0.0f;
        moff[j] = id ? (r * W_ + k) : 0;
    }

    // ---- Per-(lane,slot) channel descriptors (t-invariant) ----
    // typ: 0=cov 1=corr 2=decay 3=std 4=mean_scr 5=zscore 6=mul 7=inactive
    int typ[5], ii[5], jj[5], m0a[5];
#pragma unroll
    for (int idx = 0; idx < 5; ++idx) {
        int c = lane + 32 * idx;
        int ty = 7, i0 = 0, j0 = 0, m0 = 0;
        if (c < CTOT) {
            if (c < 36)       { ty = 0; i0 = pi[c];       j0 = pj[c]; }
            else if (c < 72)  { ty = 1; i0 = pi[c - 36];  j0 = pj[c - 36]; }
            else if (c < 81)  { ty = 2; i0 = c - 72;      j0 = i0; }
            else if (c < 90)  { ty = 3; i0 = c - 81;      j0 = i0; }
            else if (c < 99)  { ty = 4; m0 = b * HT_ + (c - 90) * T_; }
            else if (c < 108) { ty = 5; i0 = c - 99;      j0 = i0; }
            else              { ty = 6; i0 = pi[c - 108]; j0 = pj[c - 108]; }
        }
        typ[idx] = ty; ii[idx] = i0; jj[idx] = j0; m0a[idx] = m0;
    }

    // OUTPUT-only: preload fused affine params per slot.
    float q0[5], q1[5], q2[5], q3[5], q4[5], q5[5];
    bool  qpos[5];
    float* ob0[5]; float* ob1[5]; float* ob2[5];
    if (!STATS) {
#pragma unroll
        for (int idx = 0; idx < 5; ++idx) {
            int c = lane + 32 * idx;
            int g = (c < CTOT) ? grp_of(c) : 0;
            q0[idx] = red[g * 8 + 0]; q1[idx] = red[g * 8 + 1];
            q2[idx] = red[g * 8 + 2]; q3[idx] = red[g * 8 + 3];
            q4[idx] = red[g * 8 + 4]; q5[idx] = red[g * 8 + 5];
            qpos[idx] = red[g * 8 + 6] > 0.5f;
            size_t ob = (size_t)b * OUTB + (size_t)c * NW_;
            ob0[idx] = out + ob;
            ob1[idx] = out + ob + 1152;
            ob2[idx] = out + ob + 2304;
        }
    }

    float wmax[5], wmin[5], wsum[5];
    float sx[5]  = {}, sxx[5]  = {};
    float smx[5] = {}, smx2[5] = {};
    float smn[5] = {}, smn2[5] = {};
    float sav[5] = {}, sav2[5] = {};

    for (int w = 0; w < NW_; ++w) {
#pragma unroll
        for (int e = 0; e < 3; ++e) {
            const int t = 3 * w + e;
            // --- Gram via f32 WMMA: branchless operand fetch + 3 steps. ---
            const float* rp = row + t * S_;
            float mval[6];
#pragma unroll
            for (int j = 0; j < 6; ++j) {
                float dv = rp[moff[j]];
                mval[j] = fmaf(fsel[j], dv, cval[j]);
            }
            v8f acc = {0.f, 0.f, 0.f, 0.f, 0.f, 0.f, 0.f, 0.f};
#pragma unroll
            for (int s = 0; s < 3; ++s) {
                v2f a; a.x = mval[2 * s]; a.y = mval[2 * s + 1];
                acc = __builtin_amdgcn_wmma_f32_16x16x4_f32(
                    false, a, false, a, (short)0, acc, false, false);
            }
            __syncthreads();  // previous-iteration G readers done
#pragma unroll
            for (int v = 0; v < 8; ++v) G[(hv * 8 + v) * 16 + r] = acc[v];
            __syncthreads();

#pragma unroll
            for (int idx = 0; idx < 5; ++idx) {
                const int ty = typ[idx];
                const int ib = ii[idx] * 16, jb = jj[idx] * 16;
                float dot = G[ib + jj[idx]];
                float si  = G[ib + 10],      sj  = G[jb + 10];
                float dii = G[ib + ii[idx]], djj = G[jb + jj[idx]];
                float dec = G[ib + 9];
                float mi = si * 0.1f, mj = sj * 0.1f;
                float vi = fmaxf(dii * 0.1f - mi * mi, 0.0f);
                float vj = fmaxf(djj * 0.1f - mj * mj, 0.0f);
                float stdi = fsqrt_hw(vi), stdj = fsqrt_hw(vj);
                float cov = (dot - 10.0f * mi * mj) * (1.0f / 9.0f);

                float val = (ty == 0) ? cov
                          : (ty == 1) ? cov * frcp_hw(stdi * stdj + 0.01f)
                          : (ty == 2) ? dec
                          : (ty == 3) ? stdi
                          : (ty == 5) ? mi * frcp_hw(stdi + 0.01f)
                          : (ty == 6) ? dot * 0.1f
                          : 0.0f;
                if (ty == 4) {  // mean_scr: permuted gather from mean table
                    int m   = m0a[idx] + t;
                    int tt  = m / (B_ * H_);
                    int rem = m - tt * (B_ * H_);
                    int bb  = rem / H_;
                    int hh  = rem - bb * H_;
                    val = meantab[bb * HT_ + hh * T_ + tt];
                }

                if (e == 0) {
                    wmax[idx] = val; wmin[idx] = val; wsum[idx] = val;
                } else {
                    wmax[idx] = fmaxf(wmax[idx], val);
                    wmin[idx] = fminf(wmin[idx], val);
                    wsum[idx] += val;
                }
                if (STATS) { sx[idx] += val; sxx[idx] += val * val; }
                if (e == 2) {
                    if (STATS) {
                        smx[idx] += wmax[idx]; smx2[idx] += wmax[idx] * wmax[idx];
                        smn[idx] += wmin[idx]; smn2[idx] += wmin[idx] * wmin[idx];
                        float av = wsum[idx] * (1.0f / 3.0f);
                        sav[idx] += av; sav2[idx] += av * av;
                    } else if (ty != 7) {
                        float selmax = qpos[idx] ? wmax[idx] : wmin[idx];
                        float selmin = qpos[idx] ? wmin[idx] : wmax[idx];
                        ob0[idx][w] = q0[idx] * selmax    + q1[idx];
                        ob1[idx][w] = q2[idx] * wsum[idx] + q3[idx];
                        ob2[idx][w] = q4[idx] * selmin    + q5[idx];
                    }
                }
            }
        }
    }

    if (STATS) {
        __syncthreads();
#pragma unroll
        for (int idx = 0; idx < 5; ++idx) {
            int c = lane + 32 * idx;
            if (c < CTOT) {
                int g = grp_of(c);
                atomicAdd(&red[g * 8 + 0], sx[idx]);
                atomicAdd(&red[g * 8 + 1], sxx[idx]);
                atomicAdd(&red[g * 8 + 2], smx[idx]);
                atomicAdd(&red[g * 8 + 3], smx2[idx]);
                atomicAdd(&red[g * 8 + 4], smn[idx]);
                atomicAdd(&red[g * 8 + 5], smn2[idx]);
                atomicAdd(&red[g * 8 + 6], sav[idx]);
                atomicAdd(&red[g * 8 + 7], sav2[idx]);
            }
        }
        __syncthreads();
        for (int i = lane; i < 56; i += 32) atomicAdd(&gacc[i], red[i]);
    }
}

// ---------------- Pass 2: fold both BN layers into affine params -----------
__global__ void finalize_kernel(const float* __restrict__ gacc,
                                const float* __restrict__ gamma,
                                const float* __restrict__ beta,
                                float* __restrict__ params) {
    if (threadIdx.x != 0 || blockIdx.x != 0) return;
    const int chg[7] = {36, 36, 9, 9, 9, 9, 36};
    float scf[7], shf[7]; int pos[7];
    double Smx = 0, Smx2 = 0, Smn = 0, Smn2 = 0, Sav = 0, Sav2 = 0;
    for (int g = 0; g < 7; ++g) {
        double N1  = (double)B_ * chg[g] * T_;
        double mu  = (double)gacc[g * 8 + 0] / N1;
        double var = (double)gacc[g * 8 + 1] / N1 - mu * mu;
        double inv = 1.0 / sqrt(var + 1e-5);
        double s   = (double)gamma[g] * inv;
        double b0  = (double)beta[g] - mu * s;
        scf[g] = (float)s; shf[g] = (float)b0; pos[g] = (s >= 0.0);
        double Nw  = (double)B_ * chg[g] * NW_;
        double SMa = gacc[g * 8 + 2], SMa2 = gacc[g * 8 + 3];
        double SMi = gacc[g * 8 + 4], SMi2 = gacc[g * 8 + 5];
        double SAv = gacc[g * 8 + 6], SAv2 = gacc[g * 8 + 7];
        double SA  = pos[g] ? SMa  : SMi,  SA2 = pos[g] ? SMa2 : SMi2;
        double SB  = pos[g] ? SMi  : SMa,  SB2 = pos[g] ? SMi2 : SMa2;
        Smx += s * SA + b0 * Nw;  Smx2 += s * s * SA2 + 2 * s * b0 * SA + b0 * b0 * Nw;
        Smn += s * SB + b0 * Nw;  Smn2 += s * s * SB2 + 2 * s * b0 * SB + b0 * b0 * Nw;
        Sav += s * SAv + b0 * Nw; Sav2 += s * s * SAv2 + 2 * s * b0 * SAv + b0 * b0 * Nw;
    }
    double N2 = (double)B_ * CTOT * NW_;
    double s2[3], b2[3];
    double SS[3]  = {Smx, Sav, Smn};
    double SS2[3] = {Smx2, Sav2, Smn2};
    for (int k = 0; k < 3; ++k) {
        double mu  = SS[k] / N2;
        double var = SS2[k] / N2 - mu * mu;
        double inv = 1.0 / sqrt(var + 1e-5);
        s2[k] = (double)gamma[7 + k] * inv;
        b2[k] = (double)beta[7 + k] - mu * s2[k];
    }
    for (int g = 0; g < 7; ++g) {
        params[g * 8 + 0] = (float)(s2[0] * scf[g]);
        params[g * 8 + 1] = (float)(s2[0] * shf[g] + b2[0]);
        params[g * 8 + 2] = (float)(s2[1] * scf[g] / 3.0);
        params[g * 8 + 3] = (float)(s2[1] * shf[g] + b2[1]);
        params[g * 8 + 4] = (float)(s2[2] * scf[g]);
        params[g * 8 + 5] = (float)(s2[2] * shf[g] + b2[2]);
        params[g * 8 + 6] = pos[g] ? 1.0f : 0.0f;
        params[g * 8 + 7] = 0.0f;
    }
}

extern "C" void kernel_launch(void* const* d_in, const int* in_sizes, int n_in,
                              void* d_out, int out_size, void* d_ws, size_t ws_size,
                              hipStream_t stream) {
    const float* data1 = (const float*)d_in[0];
    const float* gamma = (const float*)d_in[1];
    const float* beta  = (const float*)d_in[2];
    const int*   pi    = (const int*)d_in[3];   // num
    const int*   pj    = (const int*)d_in[4];   // num_rev
    float* out = (float*)d_out;

    float* meantab = (float*)d_ws;               // B*H*T floats (~7 MB)
    float* gacc    = meantab + (size_t)B_ * HT_; // 64 floats
    float* params  = gacc + 64;                  // 64 floats

    mean_kernel<<<(B_ * HT_ + 255) / 256, 256, 0, stream>>>(data1, meantab, gacc);
    feat_kernel<true><<<B_, 32, 0, stream>>>(data1, pi, pj, meantab,
                                             gacc, nullptr, nullptr);
    finalize_kernel<<<1, 32, 0, stream>>>(gacc, gamma, beta, params);
    feat_kernel<false><<<B_, 32, 0, stream>>>(data1, pi, pj, meantab,
                                              nullptr, params, out);
}